// DynamicKGating_63857573757339
// MI455X (gfx1250) — compile-verified
//
#include <hip/hip_runtime.h>

typedef __attribute__((ext_vector_type(2))) float v2f;
typedef __attribute__((ext_vector_type(4))) float v4f;
typedef __attribute__((ext_vector_type(8))) float v8f;
typedef __attribute__((ext_vector_type(4))) unsigned int v4u;
typedef __attribute__((ext_vector_type(8))) int v8i;
typedef __attribute__((ext_vector_type(4))) int v4i;

#define TOKENS      32768
#define DMODEL      1024
#define NEXPERT     64
#define KMAX        8
#define CAP         80
#define TAU_F       0.7f
#define NFLAT       (TOKENS * KMAX)     /* 262144 */
#define CHUNK       1024
#define NCHUNK      (NFLAT / CHUNK)     /* 256 */
#define DISPATCH_SZ 167772160L          /* 8*4096*64*80 */

// ---------------------------------------------------------------------------
// Kernel 1: zero-fill dispatch+combine (1.34 GB) with NT b128 stores
// ---------------------------------------------------------------------------
__global__ void zero_fill_kernel(float* __restrict__ p, long n4) {
    long i = (long)blockIdx.x * blockDim.x + threadIdx.x;
    long stride = (long)gridDim.x * blockDim.x;
    v4f z = {0.f, 0.f, 0.f, 0.f};
    v4f* p4 = (v4f*)p;
    for (; i < n4; i += stride)
        __builtin_nontemporal_store(z, &p4[i]);   // stream: 1.34GB >> L2
}

// ---------------------------------------------------------------------------
// TDM helper: build D# groups 0/1 per ISA 8.3/8.4 and issue tensor_load_to_lds
//   group0: count=1 | lds_addr | global_addr[56:0] | type=2
//   group1: data_size/pad cfg | tensor_dim0/1 | tile_dim0/1 | dim0 stride
//   groups 2/3 (and 5th SGPR group): zero (<=2D tensor)
// ---------------------------------------------------------------------------
__device__ __forceinline__ void tdm_load_2d(
    unsigned lds_addr, unsigned long long gaddr, unsigned cfg_flags,
    unsigned tensor_d0, unsigned tensor_d1,
    unsigned tile_d0, unsigned tile_d1, unsigned long long stride0)
{
    v4u g0;
    g0.x = 1u;                                         // count=1, user mode
    g0.y = lds_addr;                                   // LDS byte address
    g0.z = (unsigned)(gaddr & 0xffffffffu);            // global_addr lo
    g0.w = (unsigned)((gaddr >> 32) & 0x1ffffffu) | (2u << 30); // hi | type=2
    v8i g1;
    g1[0] = (int)cfg_flags;                            // data_size=4B (+pad)
    g1[1] = (int)((tensor_d0 & 0xffffu) << 16);        // tensor_dim0[15:0]
    g1[2] = (int)((tensor_d0 >> 16) | ((tensor_d1 & 0xffffu) << 16));
    g1[3] = (int)((tensor_d1 >> 16) | (tile_d0 << 16)); // tile_dim0
    g1[4] = (int)(tile_d1 & 0xffffu);                  // tile_dim1 (0 = 1D)
    g1[5] = (int)(stride0 & 0xffffffffu);              // tensor_dim0_stride
    g1[6] = (int)((stride0 >> 32) & 0xffffu);
    g1[7] = 0;
    v4i z4 = {0, 0, 0, 0};
    v8i z8 = {0, 0, 0, 0, 0, 0, 0, 0};
    __builtin_amdgcn_tensor_load_to_lds(g0, g1, z4, z4, z8, 0);
}

#define CFG_F32        (2u << 16)                         /* data_size=4B   */
#define CFG_F32_PAD68  (CFG_F32 | (1u<<20) | (5u<<22) | (3u<<25))
/* pad_enable, pad_interval=5 (every 64 DWORDs), pad_amount=3 (4 DWORDs):
   produces LDS rows of 68 floats -> conflict-free A-frag ds reads          */

// ---------------------------------------------------------------------------
// Kernel 2: gating GEMM (fp32 WMMA) + softmax + dynamic top-k selection
// block = 128 threads (4 waves), 64 tokens per block
// TDM double-buffers both tiles while the waves run WMMA on current chunk.
// ---------------------------------------------------------------------------
__global__ __launch_bounds__(128) void gating_kernel(
    const float* __restrict__ x, const float* __restrict__ w,
    int* __restrict__ selIdx, float* __restrict__ selP)
{
    __shared__ __align__(16) float xs[2][64 * 68];   // x tile (TDM-padded rows)
    __shared__ __align__(16) float ws[2][64 * 64];   // W k-slab / later logits

    const int tid  = threadIdx.x;
    const int lane = tid & 31;
    const int wv   = tid >> 5;          // wave 0..3 -> tokens wv*16..+15
    const int m    = lane & 15;
    const int hi   = (lane >= 16) ? 1 : 0;
    const int tokBase = blockIdx.x * 64;

    v8f acc0 = {}; v8f acc1 = {}; v8f acc2 = {}; v8f acc3 = {};

    // preload chunks 0 and 1 (wave 0 issues; TDM ignores EXEC, in-order/wave)
    if (tid < 32) {
        #pragma unroll
        for (int c = 0; c < 2; ++c) {
            tdm_load_2d((unsigned)(uintptr_t)&xs[c][0],
                        (unsigned long long)(uintptr_t)(x + (size_t)tokBase * DMODEL + c * 64),
                        CFG_F32_PAD68, DMODEL, TOKENS, 64, 64, DMODEL);
            tdm_load_2d((unsigned)(uintptr_t)&ws[c][0],
                        (unsigned long long)(uintptr_t)(w + (size_t)c * 4096),
                        CFG_F32, DMODEL * NEXPERT, 1, 4096, 0, DMODEL * NEXPERT);
        }
    }

    for (int kc = 0; kc < 16; ++kc) {          // 64-wide k chunks
        // wait for chunk kc's two tensor ops (next chunk's two may be in flight)
        if (kc < 15) __builtin_amdgcn_s_wait_tensorcnt(2);
        else         __builtin_amdgcn_s_wait_tensorcnt(0);
        __syncthreads();

        const int b = kc & 1;
        const float* xb = &xs[b][0];
        const float* wb = &ws[b][0];
        for (int k4 = 0; k4 < 16; ++k4) {      // 4-wide K per WMMA
            const int kk = (k4 << 2) + (hi << 1);   // K0/K1 (low lanes), K2/K3 (high)
            // A frag: lane holds token row m, K=kk,kk+1
            v2f a = *reinterpret_cast<const v2f*>(&xb[(wv * 16 + m) * 68 + kk]);
            // B frags: v0 = K row kk, v1 = K row kk+1, col = expert
            v2f b0, b1, b2, b3;
            b0.x = wb[kk * 64 +      m]; b0.y = wb[(kk + 1) * 64 +      m];
            b1.x = wb[kk * 64 + 16 + m]; b1.y = wb[(kk + 1) * 64 + 16 + m];
            b2.x = wb[kk * 64 + 32 + m]; b2.y = wb[(kk + 1) * 64 + 32 + m];
            b3.x = wb[kk * 64 + 48 + m]; b3.y = wb[(kk + 1) * 64 + 48 + m];
            acc0 = __builtin_amdgcn_wmma_f32_16x16x4_f32(false, a, false, b0, (short)0, acc0, false, false);
            acc1 = __builtin_amdgcn_wmma_f32_16x16x4_f32(false, a, false, b1, (short)0, acc1, false, false);
            acc2 = __builtin_amdgcn_wmma_f32_16x16x4_f32(false, a, false, b2, (short)0, acc2, false, false);
            acc3 = __builtin_amdgcn_wmma_f32_16x16x4_f32(false, a, false, b3, (short)0, acc3, false, false);
        }
        __syncthreads();                        // all waves done reading buf b
        if (tid < 32 && kc + 2 < 16) {          // DMA chunk kc+2 into buf b
            const int nc = kc + 2;
            tdm_load_2d((unsigned)(uintptr_t)&xs[b][0],
                        (unsigned long long)(uintptr_t)(x + (size_t)tokBase * DMODEL + nc * 64),
                        CFG_F32_PAD68, DMODEL, TOKENS, 64, 64, DMODEL);
            tdm_load_2d((unsigned)(uintptr_t)&ws[b][0],
                        (unsigned long long)(uintptr_t)(w + (size_t)nc * 4096),
                        CFG_F32, DMODEL * NEXPERT, 1, 4096, 0, DMODEL * NEXPERT);
        }
    }

    // dump logits to LDS: C-layout VGPR r -> token M=r (lanes0-15) / M=r+8 (hi)
    float* lg = &ws[0][0];
    {
        int tok = wv * 16 + hi * 8;
        #pragma unroll
        for (int r = 0; r < 8; ++r) {
            lg[(tok + r) * 64 + m     ] = acc0[r];
            lg[(tok + r) * 64 + m + 16] = acc1[r];
            lg[(tok + r) * 64 + m + 32] = acc2[r];
            lg[(tok + r) * 64 + m + 48] = acc3[r];
        }
    }
    __syncthreads();
    // one thread per token: softmax + dynamic top-k
    if (tid < 64) {
        float* L = &lg[tid * 64];
        float mx = L[0];
        for (int e = 1; e < 64; ++e) mx = fmaxf(mx, L[e]);
        float s = 0.f;
        for (int e = 0; e < 64; ++e) { float p = __expf(L[e] - mx); L[e] = p; s += p; }
        float inv = 1.f / s;
        float selp[8]; int seli[8];
        for (int j = 0; j < 8; ++j) {          // iterative argmax = sorted top-8
            float best = -1.f; int bi = 0;
            for (int e = 0; e < 64; ++e) { float v = L[e]; if (v > best) { best = v; bi = e; } }
            seli[j] = bi; selp[j] = best * inv; L[bi] = -2.f;
        }
        float csum = 0.f; int cnt = 0;
        for (int j = 0; j < 8; ++j) { csum += selp[j]; cnt += (csum < TAU_F) ? 1 : 0; }
        int kstar = cnt < 1 ? 1 : (cnt > 8 ? 8 : cnt);
        float ssum = 0.f;
        for (int j = 0; j < kstar; ++j) ssum += selp[j];
        float denom = fmaxf(ssum, 1e-7f);
        int gtok = tokBase + tid;
        for (int j = 0; j < 8; ++j) {
            selIdx[gtok * 8 + j] = (j < kstar) ? seli[j] : -1;
            selP  [gtok * 8 + j] = (j < kstar) ? selp[j] / denom : 0.f;
        }
    }
}

// ---------------------------------------------------------------------------
// Kernel 3: per-chunk per-expert histogram (LDS atomics)
// ---------------------------------------------------------------------------
__global__ __launch_bounds__(64) void count_kernel(
    const int* __restrict__ selIdx, int* __restrict__ counts)
{
    __shared__ int cnt[NEXPERT];
    int tid = threadIdx.x;
    cnt[tid] = 0;
    __syncthreads();
    int base = blockIdx.x * CHUNK;
    for (int i = tid; i < CHUNK; i += 64) {
        int e = selIdx[base + i];
        if (e >= 0) atomicAdd(&cnt[e], 1);
    }
    __syncthreads();
    counts[blockIdx.x * NEXPERT + tid] = cnt[tid];
}

// ---------------------------------------------------------------------------
// Kernel 4: per-expert exclusive prefix over chunks + usage = min(T, cap)
// ---------------------------------------------------------------------------
__global__ __launch_bounds__(64) void prefix_kernel(
    const int* __restrict__ counts, int* __restrict__ offs, int* __restrict__ usage)
{
    int t = threadIdx.x;
    int run = 0;
    for (int c = 0; c < NCHUNK; ++c) {
        int v = counts[c * NEXPERT + t];
        offs[c * NEXPERT + t] = run;
        run += v;
    }
    usage[t] = (run < CAP) ? run : CAP;
}

// ---------------------------------------------------------------------------
// Kernel 5: ordered claim + scatter (thread t owns expert t)
// ---------------------------------------------------------------------------
__global__ __launch_bounds__(64) void scatter_kernel(
    const int* __restrict__ selIdx, const float* __restrict__ selP,
    const int* __restrict__ offs, float* __restrict__ out)
{
    __shared__ int ec[CHUNK];
    int tid  = threadIdx.x;
    int base = blockIdx.x * CHUNK;
    for (int i = tid; i < CHUNK; i += 64) ec[i] = selIdx[base + i];
    __syncthreads();
    int c = offs[blockIdx.x * NEXPERT + tid];
    for (int i = 0; i < CHUNK; ++i) {          // flat order = reference order
        if (ec[i] == tid) {
            if (c < CAP) {
                int n   = base + i;
                int tok = n >> 3;               // n = tok*K + k
                long idx = ((long)tok * NEXPERT + tid) * CAP + c;
                out[idx] = 1.0f;                       // dispatch
                out[DISPATCH_SZ + idx] = selP[n];      // combine
            }
            ++c;
        }
    }
}

// ---------------------------------------------------------------------------
// Kernel 6: load-balance loss scalar
// ---------------------------------------------------------------------------
__global__ __launch_bounds__(64) void loss_kernel(
    const int* __restrict__ usage, float* __restrict__ out)
{
    __shared__ float su[NEXPERT];
    su[threadIdx.x] = (float)usage[threadIdx.x];
    __syncthreads();
    if (threadIdx.x == 0) {
        float s = 0.f;
        for (int i = 0; i < NEXPERT; ++i) s += su[i];
        float mean = s / (float)NEXPERT;
        float d = 0.f;
        for (int i = 0; i < NEXPERT; ++i) { float t = su[i] - mean; d += t * t; }
        d /= (float)NEXPERT;
        out[2 * DISPATCH_SZ] = (s > 0.f) ? d / (mean + 1e-8f) : 0.f;
    }
}

// ---------------------------------------------------------------------------
extern "C" void kernel_launch(void* const* d_in, const int* in_sizes, int n_in,
                              void* d_out, int out_size, void* d_ws, size_t ws_size,
                              hipStream_t stream) {
    const float* x = (const float*)d_in[0];        // [8,4096,1024]
    const float* w = (const float*)d_in[1];        // [1024,64]
    float* out = (float*)d_out;                    // dispatch | combine | loss

    int*   selIdx = (int*)d_ws;                                        // 1 MB
    float* selP   = (float*)((char*)d_ws + (size_t)NFLAT * 4);         // 1 MB
    int*   counts = (int*)((char*)d_ws + (size_t)NFLAT * 8);           // 64 KB
    int*   offs   = counts + NCHUNK * NEXPERT;                         // 64 KB
    int*   usage  = offs + NCHUNK * NEXPERT;                           // 256 B

    // 1) zero dispatch+combine (dominant BW cost)
    long n4 = (2 * DISPATCH_SZ) / 4;
    zero_fill_kernel<<<16384, 256, 0, stream>>>(out, n4);
    // 2) gating GEMM + softmax + dynamic top-k (TDM double-buffered)
    gating_kernel<<<TOKENS / 64, 128, 0, stream>>>(x, w, selIdx, selP);
    // 3-4) global per-expert exclusive positions
    count_kernel<<<NCHUNK, 64, 0, stream>>>(selIdx, counts);
    prefix_kernel<<<1, 64, 0, stream>>>(counts, offs, usage);
    // 5) capacity-checked scatter
    scatter_kernel<<<NCHUNK, 64, 0, stream>>>(selIdx, selP, offs, out);
    // 6) loss
    loss_kernel<<<1, 64, 0, stream>>>(usage, out);
}